// scikit_anfis_76192719831219
// MI455X (gfx1250) — compile-verified
//
#include <hip/hip_runtime.h>

namespace {

constexpr int kB      = 8192;
constexpr int kNIn    = 8;
constexpr int kNMfs   = 4;
constexpr int kNRules = 2048;
constexpr int kK      = kNIn * kNMfs;   // 32 = flattened (input, mf) axis

typedef float v2f __attribute__((ext_vector_type(2)));
typedef float v8f __attribute__((ext_vector_type(8)));

// ---------------------------------------------------------------------------
// Kernel 1: L[b,k] = log2(max(x[b,k], 1e-30))   (x is contiguous (B, 32))
// Clamp avoids -inf so zero entries of S can't produce 0 * -inf = NaN.
// __builtin_amdgcn_logf -> v_log_f32 (base-2 log).
// ---------------------------------------------------------------------------
__global__ void log2_kernel(const float* __restrict__ x,
                            float* __restrict__ L, int n) {
  int i = blockIdx.x * blockDim.x + threadIdx.x;
  if (i < n) L[i] = __builtin_amdgcn_logf(fmaxf(x[i], 1e-30f));
}

// ---------------------------------------------------------------------------
// Kernel 2: one-hot selection matrix S, row-major (kNRules x 32).
// S[r][i*4+m] = 1 iff mf[r][i] == m; masked (-1) -> all-zero group (factor 1).
// Every element written each call (no reliance on workspace contents).
// ---------------------------------------------------------------------------
__global__ void build_sel_kernel(const int* __restrict__ mf,
                                 float* __restrict__ S) {
  int r = blockIdx.x * blockDim.x + threadIdx.x;
  if (r >= kNRules) return;
  float s[kK];
#pragma unroll
  for (int k = 0; k < kK; ++k) s[k] = 0.0f;
#pragma unroll
  for (int i = 0; i < kNIn; ++i) {
    int idx = mf[r * kNIn + i];
    if (idx >= 0) s[i * kNMfs + idx] = 1.0f;
  }
  float4* dst = (float4*)(S + (size_t)r * kK);
#pragma unroll
  for (int q = 0; q < kK / 4; ++q) dst[q] = ((const float4*)s)[q];
}

// ---------------------------------------------------------------------------
// Kernel 3: out = exp2(L(8192x32) x S^T(32x2048)) via V_WMMA_F32_16X16X4_F32.
// Block = 256 threads = 8 waves; block owns one 16-row M tile, waves stripe
// over 128 N tiles (16 iterations). A tile loaded once per wave into 8 v2f
// fragments (ISA 32-bit A 16x4 layout: lanes 0-15 M=lane, K {0,1};
// lanes 16-31 M=lane-16, K {2,3}).  B uses the mirrored layout with N=lane%16.
// ---------------------------------------------------------------------------
__global__ __launch_bounds__(256) void wmma_gemm_exp2_kernel(
    const float* __restrict__ L,     // (kB, 32)
    const float* __restrict__ S,     // (kNRules, 32)
    float* __restrict__ out) {       // (kB, kNRules)
  const int wave   = threadIdx.x >> 5;   // 0..7
  const int lane   = threadIdx.x & 31;
  const int laneLo = lane & 15;          // M (for A/D) or N (for B/D)
  const int laneHi = lane >> 4;          // selects K pair / M half
  const int tileM  = blockIdx.x;         // 0..511

  // A fragments for all 8 K-steps (K = 4*s + 2*laneHi + {0,1})
  const float* Arow = L + ((size_t)(tileM * 16 + laneLo)) * kK + 2 * laneHi;
  v2f a[8];
#pragma unroll
  for (int s = 0; s < 8; ++s) a[s] = *(const v2f*)(Arow + 4 * s);

  for (int t = 0; t < 16; ++t) {
    const int tileN = (t * 8 + wave) * 16;
    const float* Brow = S + ((size_t)(tileN + laneLo)) * kK + 2 * laneHi;
    v8f c = {0.f, 0.f, 0.f, 0.f, 0.f, 0.f, 0.f, 0.f};
#pragma unroll
    for (int s = 0; s < 8; ++s) {
      v2f b = *(const v2f*)(Brow + 4 * s);
      // D = A(16x4) * B(4x16) + C ; 8-arg form per CDNA5_HIP.md
      c = __builtin_amdgcn_wmma_f32_16x16x4_f32(false, a[s], false, b,
                                                (short)0, c, false, false);
    }
    // C/D layout: VGPR j -> M = j + 8*laneHi, N = laneLo
    float* orow = out + ((size_t)(tileM * 16 + laneHi * 8)) * kNRules +
                  tileN + laneLo;
#pragma unroll
    for (int j = 0; j < 8; ++j)
      orow[(size_t)j * kNRules] = __builtin_amdgcn_exp2f(c[j]);
  }
}

// ---------------------------------------------------------------------------
// Fallback (only if workspace is too small): direct gather-product.
// ---------------------------------------------------------------------------
__global__ void direct_kernel(const float* __restrict__ x,
                              const int* __restrict__ mf,
                              float* __restrict__ out) {
  int r = blockIdx.x * blockDim.x + threadIdx.x;   // rule, fastest -> coalesced
  int b = blockIdx.y;
  if (r >= kNRules) return;
  const float* xb = x + (size_t)b * kK;
  float p = 1.0f;
#pragma unroll
  for (int i = 0; i < kNIn; ++i) {
    int idx = mf[r * kNIn + i];
    if (idx >= 0) p *= xb[i * kNMfs + idx];
  }
  out[(size_t)b * kNRules + r] = p;
}

}  // namespace

extern "C" void kernel_launch(void* const* d_in, const int* in_sizes, int n_in,
                              void* d_out, int out_size, void* d_ws,
                              size_t ws_size, hipStream_t stream) {
  const float* x  = (const float*)d_in[0];
  const int*   mf = (const int*)d_in[1];
  float*       out = (float*)d_out;

  const size_t bytesL = (size_t)kB * kK * sizeof(float);        // 1 MB
  const size_t bytesS = (size_t)kNRules * kK * sizeof(float);   // 256 KB

  if (ws_size < bytesL + bytesS) {
    // Workspace too small: direct memory-bound gather-product.
    dim3 grid((kNRules + 255) / 256, kB);
    hipLaunchKernelGGL(direct_kernel, grid, dim3(256), 0, stream, x, mf, out);
    return;
  }

  float* L = (float*)d_ws;
  float* S = (float*)((char*)d_ws + bytesL);

  const int n = kB * kK;
  hipLaunchKernelGGL(log2_kernel, dim3((n + 255) / 256), dim3(256), 0, stream,
                     x, L, n);
  hipLaunchKernelGGL(build_sel_kernel, dim3((kNRules + 255) / 256), dim3(256),
                     0, stream, mf, S);
  hipLaunchKernelGGL(wmma_gemm_exp2_kernel, dim3(kB / 16), dim3(256), 0,
                     stream, L, S, out);
}